// Attention_59030030516520
// MI455X (gfx1250) — compile-verified
//
#include <hip/hip_runtime.h>
#include <math.h>

// ---------------------------------------------------------------------------
// CDNA5 (gfx1250) fused attention, fp32 WMMA (V_WMMA_F32_16X16X4_F32).
// B=2, N=2048, C=1024, H=16, D=64.
// Round 3: 4x2 register-blocked WMMA GEMMs (0.75 fragment loads per WMMA),
// double-buffered async global->LDS staging (ASYNCcnt latency hiding) in the
// flash-attention kernel, WGP-scope prefetch in the GEMM streams.
// ---------------------------------------------------------------------------

typedef __attribute__((ext_vector_type(2))) float v2f;
typedef __attribute__((ext_vector_type(8))) float v8f;

union V8 { v8f v; float f[8]; };

__device__ __forceinline__ v8f wmma_f32(v2f a, v2f b, v8f c) {
  // 8 args: (neg_a, A, neg_b, B, c_mod, C, reuse_a, reuse_b)
  return __builtin_amdgcn_wmma_f32_16x16x4_f32(false, a, false, b, (short)0, c,
                                               false, false);
}

#define BATCH   2
#define SEQ     2048
#define CDIM    1024
#define HEADS   16
#define HDIM    64
#define MROWS   (BATCH * SEQ)          // 4096
#define BHN     (BATCH * HEADS * SEQ)  // 65536 rows of q (and of k)
#define ELEMS   (BATCH * HEADS * SEQ * HDIM)  // 4194304 floats per tensor

// ---------------------------------------------------------------------------
// 4x2 register-blocked fp32-WMMA GEMM mainloop: 64x32 output per wave.
// Lane decomposition (ISA 16x4 fp32 A layout): lo = lane&15 selects the
// M-row (A) / N-row of W (B); h2 = lane>>4 selects k-subpair {2h2, 2h2+1}.
// 24 b64 fragment loads feed 32 WMMAs per 16-k step.
// X points at row (tm*64 + lo); W points at row (tn*32 + lo).
// ---------------------------------------------------------------------------
__device__ __forceinline__ void gemm_tile4x2(const float* __restrict__ X,
                                             const float* __restrict__ W,
                                             int K, int kshift, V8 acc[4][2]) {
  for (int k = 0; k < K; k += 16) {
    __builtin_prefetch((const void*)(X + k + 256), 0, 3);  // WGP-scope
    __builtin_prefetch((const void*)(W + k + 256), 0, 3);
    v2f a[4][4], b[2][4];
#pragma unroll
    for (int j = 0; j < 4; ++j) {
#pragma unroll
      for (int mi = 0; mi < 4; ++mi)
        a[mi][j] = *(const v2f*)(X + (size_t)mi * 16 * CDIM + k + 4 * j + kshift);
#pragma unroll
      for (int ni = 0; ni < 2; ++ni)
        b[ni][j] = *(const v2f*)(W + (size_t)ni * 16 * CDIM + k + 4 * j + kshift);
    }
#pragma unroll
    for (int j = 0; j < 4; ++j)
#pragma unroll
      for (int mi = 0; mi < 4; ++mi)
#pragma unroll
        for (int ni = 0; ni < 2; ++ni)
          acc[mi][ni].v = wmma_f32(a[mi][j], b[ni][j], acc[mi][ni].v);
  }
}

// ---------------------------------------------------------------------------
// Kernel 1: QKV projection. y[4096,3072] = x @ qkv_w^T + qkv_b, scattered
// into q/k/v with [B,H,N,D] layout. 4 waves/block, one 64x32 macro-tile/wave.
// ---------------------------------------------------------------------------
__global__ __launch_bounds__(128) void k_qkv(const float* __restrict__ x,
                                             const float* __restrict__ w,
                                             const float* __restrict__ bias,
                                             float* __restrict__ q,
                                             float* __restrict__ kk,
                                             float* __restrict__ v) {
  const int lane = threadIdx.x & 31, wid = threadIdx.x >> 5;
  const int lo = lane & 15, h2 = lane >> 4;
  const int TN = (3 * CDIM) / 32;  // 96 col macro-tiles
  int tile = blockIdx.x * 4 + wid;
  int tm = tile / TN, tn = tile % TN;

  V8 acc[4][2];
#pragma unroll
  for (int mi = 0; mi < 4; ++mi)
#pragma unroll
    for (int ni = 0; ni < 2; ++ni)
#pragma unroll
      for (int r = 0; r < 8; ++r) acc[mi][ni].f[r] = 0.0f;

  gemm_tile4x2(x + (size_t)(tm * 64 + lo) * CDIM,
               w + (size_t)(tn * 32 + lo) * CDIM, CDIM, 2 * h2, acc);

#pragma unroll
  for (int ni = 0; ni < 2; ++ni) {
    int col = tn * 32 + ni * 16 + lo;  // 0..3071
    float bv = bias[col];
    int which = col >> 10;             // 0:q 1:k 2:v
    int cc = col & (CDIM - 1);
    int hh = cc >> 6, dd = cc & 63;
    float* dst = (which == 0) ? q : ((which == 1) ? kk : v);
#pragma unroll
    for (int mi = 0; mi < 4; ++mi) {
#pragma unroll
      for (int r = 0; r < 8; ++r) {
        int row = tm * 64 + mi * 16 + r + 8 * h2;  // 0..4095
        int b = row >> 11, nn = row & (SEQ - 1);
        dst[(((size_t)b * HEADS + hh) * SEQ + nn) * HDIM + dd] =
            acc[mi][ni].f[r] + bv;
      }
    }
  }
}

// ---------------------------------------------------------------------------
// Kernel 2: per-head RMSNorm + segmented RoPE, in place on q and k.
// One wave per (b,h,n) row of D=64; lane i owns the RoPE pair (i, i+32).
// ---------------------------------------------------------------------------
__global__ __launch_bounds__(256) void k_normrope(float* __restrict__ q,
                                                  float* __restrict__ kk,
                                                  const float* __restrict__ qw,
                                                  const float* __restrict__ kw) {
  const int lane = threadIdx.x & 31, wid = threadIdx.x >> 5;
  size_t row = (size_t)blockIdx.x * 8 + wid;   // 0..131071
  float* t;
  const float* w;
  size_t rr;
  if (row < (size_t)BHN) { t = q;  w = qw; rr = row; }
  else                   { t = kk; w = kw; rr = row - BHN; }
  float* base = t + rr * HDIM;

  float v1 = base[lane], v2 = base[lane + 32];
  float ss = v1 * v1 + v2 * v2;
#pragma unroll
  for (int off = 16; off; off >>= 1) ss += __shfl_xor(ss, off, 32);
  float scale = rsqrtf(ss * (1.0f / 64.0f) + 1e-6f);
  v1 = v1 * scale * w[lane];
  v2 = v2 * scale * w[lane + 32];

  int nn = (int)(rr & (SEQ - 1));
  int pos = -1;                       // segments (1024, 512); tail untouched
  if (nn < 1024) pos = nn;
  else if (nn < 1536) pos = nn - 1024;

  float o1 = v1, o2 = v2;
  if (pos >= 0) {
    float inv_freq = __powf(10000.0f, -(float)lane * (1.0f / 32.0f));
    float ang = (float)pos * inv_freq;
    float s, c;
    __sincosf(ang, &s, &c);
    o1 = v1 * c - v2 * s;
    o2 = v2 * c + v1 * s;
  }
  base[lane] = o1;
  base[lane + 32] = o2;
}

// ---------------------------------------------------------------------------
// Kernel 3: flash attention per (b,h). Block = 4 waves = 64 query rows;
// loop over 128 K/V tiles of 16 rows, DOUBLE-BUFFERED through LDS with
// GLOBAL_LOAD_ASYNC_TO_LDS_B128: tile kt+1 is issued before computing on
// tile kt, and the wave only waits on ASYNCcnt at the end of the iteration,
// hiding the global->LDS latency under the 32 WMMAs + softmax.
// Online softmax: each 16x16 score row lives on one 16-lane half in one
// accumulator slot, so running max/sum state maps 1:1 onto the per-lane
// C-layout rows. Output written as [B,N,H,D] so the projection reads a
// plain [M,C] matrix.
// ---------------------------------------------------------------------------
__global__ __launch_bounds__(128) void k_attn(const float* __restrict__ q,
                                              const float* __restrict__ kk,
                                              const float* __restrict__ v,
                                              float* __restrict__ o) {
  const int lane = threadIdx.x & 31, wid = threadIdx.x >> 5;
  const int lo = lane & 15, h2 = lane >> 4;
  const int bh = blockIdx.x >> 5;              // 0..31  (b*16+h)
  const int qb = (blockIdx.x & 31) * 64;       // query block start

  __shared__ float kbuf[2][16 * 64];
  __shared__ float vbuf[2][16 * 64];
  __shared__ float sP[4][16 * 18];             // padded 16x16 P tile per wave

  const float* qbase = q  + (size_t)bh * SEQ * HDIM;
  const float* kbase = kk + (size_t)bh * SEQ * HDIM;
  const float* vbase = v  + (size_t)bh * SEQ * HDIM;

  const int qrow = qb + wid * 16 + lo;
  v2f qa[16];
#pragma unroll
  for (int t = 0; t < 16; ++t)
    qa[t] = *(const v2f*)(qbase + (size_t)qrow * HDIM + 4 * t + 2 * h2);

  V8 o0, o1, o2, o3;
  float mst[8], lst[8];
#pragma unroll
  for (int r = 0; r < 8; ++r) {
    o0.f[r] = o1.f[r] = o2.f[r] = o3.f[r] = 0.0f;
    mst[r] = -INFINITY;
    lst[r] = 0.0f;
  }
  float* sPw = sP[wid];

  // per-thread staging addresses: each thread owns 8 consecutive floats
  // (two b128 transfers) of each 16x64 tile, per buffer
  const int ti = threadIdx.x * 8;
  const unsigned klds[2] = {(unsigned)(uintptr_t)(void*)(&kbuf[0][ti]),
                            (unsigned)(uintptr_t)(void*)(&kbuf[1][ti])};
  const unsigned vlds[2] = {(unsigned)(uintptr_t)(void*)(&vbuf[0][ti]),
                            (unsigned)(uintptr_t)(void*)(&vbuf[1][ti])};

  auto issue_tile = [&](int nb, int kt) {
    const float* kg = kbase + (size_t)kt * 16 * HDIM + ti;
    const float* vg = vbase + (size_t)kt * 16 * HDIM + ti;
    asm volatile(
        "global_load_async_to_lds_b128 %0, %2, off\n\t"
        "global_load_async_to_lds_b128 %0, %2, off offset:16\n\t"
        "global_load_async_to_lds_b128 %1, %3, off\n\t"
        "global_load_async_to_lds_b128 %1, %3, off offset:16"
        :
        : "v"(klds[nb]), "v"(vlds[nb]), "v"(kg), "v"(vg)
        : "memory");
  };

  // prologue: stage tile 0 into buffer 0
  issue_tile(0, 0);
  asm volatile("s_wait_asynccnt 0x0" ::: "memory");
  __syncthreads();

  for (int kt = 0; kt < SEQ / 16; ++kt) {
    const int cur = kt & 1;
    // prefetch next K/V tile into the other buffer (consumed-before barrier
    // at the end of the previous iteration makes it safe to overwrite)
    if (kt + 1 < SEQ / 16) issue_tile(cur ^ 1, kt + 1);

    const float* kb = kbuf[cur];
    const float* vb = vbuf[cur];

    // S = (Q @ K^T) * scale  : 16 fp32 WMMAs over D=64
    V8 s;
#pragma unroll
    for (int r = 0; r < 8; ++r) s.f[r] = 0.0f;
#pragma unroll
    for (int t = 0; t < 16; ++t) {
      v2f bk = *(const v2f*)(kb + lo * HDIM + 4 * t + 2 * h2);
      s.v = wmma_f32(qa[t], bk, s.v);
    }

    // online softmax per row (row = half-wave of 16 lanes)
    float p[8];
#pragma unroll
    for (int r = 0; r < 8; ++r) {
      float sv = s.f[r] * 0.125f;  // 1/sqrt(64)
      float rm = sv;
      rm = fmaxf(rm, __shfl_xor(rm, 1, 32));
      rm = fmaxf(rm, __shfl_xor(rm, 2, 32));
      rm = fmaxf(rm, __shfl_xor(rm, 4, 32));
      rm = fmaxf(rm, __shfl_xor(rm, 8, 32));
      float nm = fmaxf(mst[r], rm);
      float alpha = __expf(mst[r] - nm);
      float pe = __expf(sv - nm);
      float rs = pe;
      rs += __shfl_xor(rs, 1, 32);
      rs += __shfl_xor(rs, 2, 32);
      rs += __shfl_xor(rs, 4, 32);
      rs += __shfl_xor(rs, 8, 32);
      lst[r] = lst[r] * alpha + rs;
      mst[r] = nm;
      o0.f[r] *= alpha;
      o1.f[r] *= alpha;
      o2.f[r] *= alpha;
      o3.f[r] *= alpha;
      p[r] = pe;
    }

    // transpose P (C/D layout -> A layout) through padded LDS
#pragma unroll
    for (int r = 0; r < 8; ++r)
      sPw[(r + 8 * h2) * 18 + lo] = p[r];
    __syncthreads();

    // O += P @ V : 16 fp32 WMMAs (4 k-steps x 4 d-blocks)
#pragma unroll
    for (int t = 0; t < 4; ++t) {
      v2f pa = *(const v2f*)(sPw + lo * 18 + 4 * t + 2 * h2);
      int krow = 4 * t + 2 * h2;
      v2f b0, b1, b2, b3;
      b0.x = vb[krow * HDIM + 0  + lo]; b0.y = vb[(krow + 1) * HDIM + 0  + lo];
      b1.x = vb[krow * HDIM + 16 + lo]; b1.y = vb[(krow + 1) * HDIM + 16 + lo];
      b2.x = vb[krow * HDIM + 32 + lo]; b2.y = vb[(krow + 1) * HDIM + 32 + lo];
      b3.x = vb[krow * HDIM + 48 + lo]; b3.y = vb[(krow + 1) * HDIM + 48 + lo];
      o0.v = wmma_f32(pa, b0, o0.v);
      o1.v = wmma_f32(pa, b1, o1.v);
      o2.v = wmma_f32(pa, b2, o2.v);
      o3.v = wmma_f32(pa, b3, o3.v);
    }

    // wait for the prefetched tile to land; fence all waves before reuse
    asm volatile("s_wait_asynccnt 0x0" ::: "memory");
    __syncthreads();
  }

  // normalize + store as [B,N,H,D]
  const int b = bh >> 4, hh = bh & 15;
#pragma unroll
  for (int r = 0; r < 8; ++r) {
    float inv = 1.0f / lst[r];
    int n = qb + wid * 16 + r + 8 * h2;
    size_t rowoff = (((size_t)b * SEQ + n) * HEADS + hh) * HDIM;
    o[rowoff + 0  + lo] = o0.f[r] * inv;
    o[rowoff + 16 + lo] = o1.f[r] * inv;
    o[rowoff + 32 + lo] = o2.f[r] * inv;
    o[rowoff + 48 + lo] = o3.f[r] * inv;
  }
}

// ---------------------------------------------------------------------------
// Kernel 4: output projection. out[4096,1024] = attn @ proj_w^T + proj_b.
// ---------------------------------------------------------------------------
__global__ __launch_bounds__(128) void k_proj(const float* __restrict__ a,
                                              const float* __restrict__ w,
                                              const float* __restrict__ bias,
                                              float* __restrict__ y) {
  const int lane = threadIdx.x & 31, wid = threadIdx.x >> 5;
  const int lo = lane & 15, h2 = lane >> 4;
  const int TN = CDIM / 32;  // 32 col macro-tiles
  int tile = blockIdx.x * 4 + wid;
  int tm = tile / TN, tn = tile % TN;

  V8 acc[4][2];
#pragma unroll
  for (int mi = 0; mi < 4; ++mi)
#pragma unroll
    for (int ni = 0; ni < 2; ++ni)
#pragma unroll
      for (int r = 0; r < 8; ++r) acc[mi][ni].f[r] = 0.0f;

  gemm_tile4x2(a + (size_t)(tm * 64 + lo) * CDIM,
               w + (size_t)(tn * 32 + lo) * CDIM, CDIM, 2 * h2, acc);

#pragma unroll
  for (int ni = 0; ni < 2; ++ni) {
    int col = tn * 32 + ni * 16 + lo;
    float bv = bias[col];
#pragma unroll
    for (int mi = 0; mi < 4; ++mi) {
#pragma unroll
      for (int r = 0; r < 8; ++r) {
        int row = tm * 64 + mi * 16 + r + 8 * h2;
        y[(size_t)row * CDIM + col] = acc[mi][ni].f[r] + bv;
      }
    }
  }
}

// ---------------------------------------------------------------------------
extern "C" void kernel_launch(void* const* d_in, const int* in_sizes, int n_in,
                              void* d_out, int out_size, void* d_ws,
                              size_t ws_size, hipStream_t stream) {
  const float* x      = (const float*)d_in[0];
  const float* qkv_w  = (const float*)d_in[1];
  const float* qkv_b  = (const float*)d_in[2];
  const float* qn_w   = (const float*)d_in[3];
  const float* kn_w   = (const float*)d_in[4];
  const float* proj_w = (const float*)d_in[5];
  const float* proj_b = (const float*)d_in[6];
  float* out = (float*)d_out;

  float* ws = (float*)d_ws;
  float* qbuf = ws;                      // [B,H,N,D]
  float* kbuf = ws + (size_t)ELEMS;      // [B,H,N,D]
  float* vbuf = ws + (size_t)ELEMS * 2;  // [B,H,N,D]
  float* abuf = ws + (size_t)ELEMS * 3;  // attention out, [B,N,H,D]

  // 1) QKV projection: (4096/64)*(3072/32) = 6144 macro-tiles, 4 per block
  k_qkv<<<(64 * 96) / 4, 128, 0, stream>>>(x, qkv_w, qkv_b, qbuf, kbuf, vbuf);

  // 2) RMSNorm + segmented RoPE on q and k: 131072 rows, 8 waves per block
  k_normrope<<<(2 * BHN) / 8, 256, 0, stream>>>(qbuf, kbuf, qn_w, kn_w);

  // 3) flash attention: B*H*(N/64) = 1024 blocks
  k_attn<<<BATCH * HEADS * (SEQ / 64), 128, 0, stream>>>(qbuf, kbuf, vbuf, abuf);

  // 4) output projection: (4096/64)*(1024/32) = 2048 macro-tiles, 4 per block
  k_proj<<<(64 * 32) / 4, 128, 0, stream>>>(abuf, proj_w, proj_b, out);
}